// GINConv_28716151341445
// MI455X (gfx1250) — compile-verified
//
#include <hip/hip_runtime.h>

// GINConv aggregation for MI455X (gfx1250, wave32).
//   out = (1+eps)*feat + segment_sum(feat[edge_src], edge_dst)
// Memory-bound (AI ~0.5 FLOP/B). feat (25.6MB) and out (25.6MB) both fit in
// the 192MB L2, so the scatter phase is bound by L2 atomic throughput, not
// HBM. Strategy: 16 lanes/edge * float4 => one 256B row per edge via
// GLOBAL_LOAD_B128, 4 no-return GLOBAL_ATOMIC_ADD_F32 per lane, edge indices
// staged once per block into LDS via the gfx1250 async global->LDS path
// (ASYNCcnt) when available, prefetch of the index stream ahead.

#define D_FEAT   64
#define EPB      16   // edges per 256-thread block (16 threads per edge)

#if defined(__gfx1250__) && __has_builtin(__builtin_amdgcn_global_load_async_to_lds_b32)
#define HAVE_ASYNC_LDS 1
#else
#define HAVE_ASYNC_LDS 0
#endif

typedef __attribute__((address_space(1))) int global_i32;
typedef __attribute__((address_space(3))) int lds_i32;

__global__ __launch_bounds__(256) void gin_init_kernel(
    const float* __restrict__ feat,
    const float* __restrict__ eps,
    float* __restrict__ out,
    int n4)
{
  int i = blockIdx.x * blockDim.x + threadIdx.x;
  if (i >= n4) return;
  const float s = 1.0f + eps[0];
  float4 f = reinterpret_cast<const float4*>(feat)[i];
  float4 o;
  o.x = s * f.x;
  o.y = s * f.y;
  o.z = s * f.z;
  o.w = s * f.w;
  reinterpret_cast<float4*>(out)[i] = o;
}

__global__ __launch_bounds__(256) void gin_scatter_kernel(
    const float* __restrict__ feat,
    const int*   __restrict__ edge_src,
    const int*   __restrict__ edge_dst,
    float*       __restrict__ out,
    int n_edges)
{
  // Stage this block's 16 src + 16 dst indices into LDS so each edge's
  // indices are read from global memory once, not 16 times. On gfx1250 use
  // GLOBAL_LOAD_ASYNC_TO_LDS_B32: memory -> LDS directly, no VGPR round
  // trip, tracked on ASYNCcnt.
  __shared__ int s_idx[2 * EPB];

  const int tid = threadIdx.x;
  const int e0  = blockIdx.x * EPB;

  if (tid < 2 * EPB) {
    const int which = tid >> 4;          // 0 -> src array, 1 -> dst array
    const int k     = tid & (EPB - 1);   // edge slot within block
    const int e     = e0 + k;
    const int* p    = which ? edge_dst : edge_src;
    if (e < n_edges) {
#if HAVE_ASYNC_LDS
      __builtin_amdgcn_global_load_async_to_lds_b32(
          (global_i32*)(uintptr_t)(p + e),
          (lds_i32*)(uint32_t)(uintptr_t)(s_idx + tid),
          /*offset=*/0, /*cpol=*/0);
#else
      s_idx[tid] = p[e];
#endif
    } else {
      s_idx[tid] = 0;
    }
    // Run the index stream ahead of the dependent gathers (global_prefetch_b8).
    __builtin_prefetch(p + e + 64 * EPB, 0, 0);
  }
#if HAVE_ASYNC_LDS
#if __has_builtin(__builtin_amdgcn_s_wait_asynccnt)
  __builtin_amdgcn_s_wait_asynccnt(0);
#else
  asm volatile("s_wait_asynccnt 0" ::: "memory");
#endif
#endif
  __syncthreads();

  const int el = tid >> 4;   // edge slot 0..15
  const int c  = tid & 15;   // float4 chunk 0..15 within the 64-float row
  const int e  = e0 + el;
  if (e >= n_edges) return;

  const int s = s_idx[el];
  const int d = s_idx[EPB + el];

  // 256B row per edge, 16B per lane: GLOBAL_LOAD_B128, fully coalesced.
  const float4 v =
      *reinterpret_cast<const float4*>(feat + (size_t)s * D_FEAT + c * 4);

  // Fire-and-forget f32 atomics resolved in L2 (no-return -> STOREcnt).
  float* o = out + (size_t)d * D_FEAT + c * 4;
  atomicAdd(o + 0, v.x);
  atomicAdd(o + 1, v.y);
  atomicAdd(o + 2, v.z);
  atomicAdd(o + 3, v.w);
}

extern "C" void kernel_launch(void* const* d_in, const int* in_sizes, int n_in,
                              void* d_out, int out_size, void* d_ws, size_t ws_size,
                              hipStream_t stream) {
  const float* feat     = (const float*)d_in[0];   // [N, 64] f32
  const float* eps      = (const float*)d_in[1];   // [1]     f32
  const int*   edge_src = (const int*)d_in[2];     // [E]     i32
  const int*   edge_dst = (const int*)d_in[3];     // [E]     i32
  float*       out      = (float*)d_out;           // [N, 64] f32

  const int nd = in_sizes[0];          // N * 64
  const int E  = in_sizes[2];          // number of edges
  const int n4 = nd / 4;               // float4 elements

  // Phase 1: out = (1+eps) * feat  (also initializes the accumulator,
  // required every call since the harness poisons d_out).
  gin_init_kernel<<<(n4 + 255) / 256, 256, 0, stream>>>(feat, eps, out, n4);

  // Phase 2: scatter-add gathered source rows into destination rows.
  gin_scatter_kernel<<<(E + EPB - 1) / EPB, 256, 0, stream>>>(
      feat, edge_src, edge_dst, out, E);
}